// GraphConv_90151363543363
// MI455X (gfx1250) — compile-verified
//
#include <hip/hip_runtime.h>

typedef __attribute__((ext_vector_type(2))) float v2f;
typedef __attribute__((ext_vector_type(8))) float v8f;

#define NN 100000
#define DD 128

// ---------------------------------------------------------------------------
// h = x @ W   (fp32, V_WMMA_F32_16X16X4_F32)
// Block = 256 threads = 8 waves. blockIdx.x selects a 16-row strip; wave w
// computes the 16x16 tile at columns [16w, 16w+16). K loop: 128/4 = 32 WMMAs.
//
// A-fragment (16x4 f32, ISA 7.12.2): lanes 0-15 hold M=lane with K={0,1} in
// v[0],v[1]; lanes 16-31 hold M=lane-16 with K={2,3}  -> one contiguous
// float2 load per lane from row-major x.
// B-fragment (4x16 f32): lanes 0-15 hold N=lane with K={0,1}; lanes 16-31
// hold N=lane-16 with K={2,3} -> two dword loads, stride 128 floats.
// C/D (16x16 f32): VGPR r -> M = r + 8*(lane>=16), N = lane&15.
// ---------------------------------------------------------------------------
__global__ __launch_bounds__(256) void gemm16(const float* __restrict__ A,
                                              const float* __restrict__ W,
                                              float* __restrict__ H) {
    const int row0 = blockIdx.x * 16;
    const int lane = threadIdx.x & 31;
    const int col0 = (threadIdx.x >> 5) * 16;
    const int mRow = lane & 15;          // A: M index
    const int kHi  = (lane >> 4) << 1;   // A/B: K sub-offset (0 or 2)
    const int n    = lane & 15;          // B/C: N index
    const int mOff = (lane >> 4) << 3;   // C: +8 rows for upper half-wave

    v8f c = {};
    const float* aPtr = A + (size_t)(row0 + mRow) * DD + kHi;
    const float* bPtr = W + (size_t)kHi * DD + (col0 + n);

#pragma unroll
    for (int k = 0; k < DD; k += 4) {
        v2f a = *(const v2f*)(aPtr + k);               // global_load_b64
        v2f b;
        b.x = bPtr[(size_t)k * DD];
        b.y = bPtr[(size_t)k * DD + DD];
        c = __builtin_amdgcn_wmma_f32_16x16x4_f32(
                /*neg_a=*/false, a, /*neg_b=*/false, b,
                /*c_mod=*/(short)0, c, /*reuse_a=*/false, /*reuse_b=*/false);
    }

#pragma unroll
    for (int r = 0; r < 8; ++r)
        H[(size_t)(row0 + mOff + r) * DD + col0 + n] = c[r];
}

// ---------------------------------------------------------------------------
// Edge scatter: one wave per edge. Each lane reads float4 of h[col] (whole
// wave reads the full 128-float row, coalesced 512B) and atomically adds it
// into agg[row]. agg (51 MB) and h (51 MB) both fit in the 192 MB L2, so the
// random gather + float atomics resolve in L2, not HBM.
// ---------------------------------------------------------------------------
__global__ __launch_bounds__(256) void edge_scatter(const int* __restrict__ ei,
                                                    const float* __restrict__ h,
                                                    float* __restrict__ agg,
                                                    float* __restrict__ deg,
                                                    int E) {
    const int e = blockIdx.x * 8 + (threadIdx.x >> 5);
    if (e >= E) return;
    const int lane = threadIdx.x & 31;
    const int r = ei[e];       // destination row (segment id)
    const int c = ei[E + e];   // source node

    float4 v = ((const float4*)(h + (size_t)c * DD))[lane];
    float* dst = agg + (size_t)r * DD + lane * 4;
    unsafeAtomicAdd(dst + 0, v.x);   // global_atomic_add_f32 (no return)
    unsafeAtomicAdd(dst + 1, v.y);
    unsafeAtomicAdd(dst + 2, v.z);
    unsafeAtomicAdd(dst + 3, v.w);
    if (lane == 0) unsafeAtomicAdd(deg + r, 1.0f);
}

// ---------------------------------------------------------------------------
// out = x @ Wroot + agg / max(deg,1) + bias   (same WMMA tiling + epilogue)
// ---------------------------------------------------------------------------
__global__ __launch_bounds__(256) void gemm_final(const float* __restrict__ X,
                                                  const float* __restrict__ RW,
                                                  const float* __restrict__ bias,
                                                  const float* __restrict__ agg,
                                                  const float* __restrict__ deg,
                                                  float* __restrict__ out) {
    const int row0 = blockIdx.x * 16;
    const int lane = threadIdx.x & 31;
    const int col0 = (threadIdx.x >> 5) * 16;
    const int mRow = lane & 15;
    const int kHi  = (lane >> 4) << 1;
    const int n    = lane & 15;
    const int mOff = (lane >> 4) << 3;

    v8f c = {};
    const float* aPtr = X + (size_t)(row0 + mRow) * DD + kHi;
    const float* bPtr = RW + (size_t)kHi * DD + (col0 + n);

#pragma unroll
    for (int k = 0; k < DD; k += 4) {
        v2f a = *(const v2f*)(aPtr + k);
        v2f b;
        b.x = bPtr[(size_t)k * DD];
        b.y = bPtr[(size_t)k * DD + DD];
        c = __builtin_amdgcn_wmma_f32_16x16x4_f32(
                false, a, false, b, (short)0, c, false, false);
    }

    const float bn = bias[col0 + n];
#pragma unroll
    for (int r = 0; r < 8; ++r) {
        const int m = row0 + mOff + r;
        const float inv = 1.0f / fmaxf(deg[m], 1.0f);
        out[(size_t)m * DD + col0 + n] =
            c[r] + agg[(size_t)m * DD + col0 + n] * inv + bn;
    }
}

extern "C" void kernel_launch(void* const* d_in, const int* in_sizes, int n_in,
                              void* d_out, int out_size, void* d_ws, size_t ws_size,
                              hipStream_t stream) {
    const float* x    = (const float*)d_in[0];
    const int*   ei   = (const int*)d_in[1];     // edge_index [2,E]
    const float* w    = (const float*)d_in[2];
    const float* rw   = (const float*)d_in[3];
    const float* bias = (const float*)d_in[4];
    float* out = (float*)d_out;

    // workspace: h [N*128] | agg [N*128] | deg [N]
    float* h   = (float*)d_ws;
    float* agg = h + (size_t)NN * DD;
    float* deg = agg + (size_t)NN * DD;
    const int E = in_sizes[1] / 2;

    // zero the accumulation buffers every call (graph-replay safe)
    hipMemsetAsync(agg, 0, ((size_t)NN * DD + NN) * sizeof(float), stream);

    gemm16<<<NN / 16, 256, 0, stream>>>(x, w, h);                 // 6250 blocks
    edge_scatter<<<(E + 7) / 8, 256, 0, stream>>>(ei, h, agg, deg, E);
    gemm_final<<<NN / 16, 256, 0, stream>>>(x, rw, bias, agg, deg, out);
}